// R2DBlock_78185584656832
// MI455X (gfx1250) — compile-verified
//
#include <hip/hip_runtime.h>
#include <stdint.h>

typedef float v4f __attribute__((ext_vector_type(4)));
typedef int   v4i __attribute__((ext_vector_type(4)));

#define AS1 __attribute__((address_space(1)))
#define AS3 __attribute__((address_space(3)))

static constexpr int BLOCK = 256;

// ---------------------------------------------------------------------------
// Per-row 2D Gram-Schmidt, matching the JAX reference numerics.
// Identity used: 1/max(sqrt(s), 1e-12) == rsqrt(max(s, 1e-24))  (monotone sqrt)
//   out row = [xh0, yh0, xh1, yh1]   (from stack([xh, yh], axis=2))
// ---------------------------------------------------------------------------
__device__ __forceinline__ v4f gram2(v4f v) {
  const float a0 = v.x, a1 = v.y, b0 = v.z, b1 = v.w;
  const float s1 = a0 * a0 + a1 * a1;
  const float i1 = rsqrtf(fmaxf(s1, 1e-24f));
  const float x0 = a0 * i1, x1 = a1 * i1;
  const float d  = x0 * b0 + x1 * b1;
  const float t0 = b0 - x0 * d, t1 = b1 - x1 * d;
  const float s2 = t0 * t0 + t1 * t1;
  const float i2 = rsqrtf(fmaxf(s2, 1e-24f));
  v4f r;
  r.x = x0;
  r.y = t0 * i2;
  r.z = x1;
  r.w = t1 * i2;
  return r;
}

// ---------------------------------------------------------------------------
// CDNA5 async global->LDS copy of one 16-byte row (ASYNCcnt-tracked).
// Lowers to: global_load_async_to_lds_b128 (GVS mode, scale_offset)
// ---------------------------------------------------------------------------
__device__ __forceinline__ void async_copy16(const v4f* gsrc, v4f* lds_dst) {
  __builtin_amdgcn_global_load_async_to_lds_b128(
      (AS1 v4i*)(uintptr_t)gsrc,
      (AS3 v4i*)(uint32_t)(uintptr_t)lds_dst,
      /*offset=*/0, /*cpol=*/0);
}

template <int N>
__device__ __forceinline__ void wait_async_le() {
  asm volatile("s_wait_asynccnt %0" ::"n"(N) : "memory");
}

// ---------------------------------------------------------------------------
// 4-deep async-pipelined streaming kernel (up to 3 outstanding loads/wave).
// Requires rows == T * iters, T == gridDim.x * BLOCK (checked on host).
// Loads complete in order (ASYNCcnt), so waiting for
//   outstanding <= min(iters-1-k, 3)
// guarantees slot[k & 3] is resident. All wait-branches are wave-uniform.
// ---------------------------------------------------------------------------
__global__ __launch_bounds__(BLOCK) void ortho2_pipe(const v4f* __restrict__ in,
                                                     v4f* __restrict__ out,
                                                     int T, int iters) {
  __shared__ v4f sh[BLOCK * 4];
  const int tid = blockIdx.x * BLOCK + threadIdx.x;
  v4f* slot = &sh[threadIdx.x * 4];

  async_copy16(&in[(size_t)tid], &slot[0]);
  if (iters > 1) async_copy16(&in[(size_t)tid + (size_t)T], &slot[1]);
  if (iters > 2) async_copy16(&in[(size_t)tid + 2 * (size_t)T], &slot[2]);

  for (int k = 0; k < iters; ++k) {
    const int cur = k & 3;
    if (k + 3 < iters) {
      async_copy16(&in[(size_t)tid + (size_t)(k + 3) * (size_t)T],
                   &slot[(cur + 3) & 3]);
      wait_async_le<3>();
    } else if (k + 2 < iters) {
      wait_async_le<2>();
    } else if (k + 1 < iters) {
      wait_async_le<1>();
    } else {
      wait_async_le<0>();
    }
    const v4f v = slot[cur];                      // ds_load_b128
    __builtin_nontemporal_store(gram2(v),         // global_store_b128 th:NT
                                &out[(size_t)tid + (size_t)k * (size_t)T]);
  }
}

// ---------------------------------------------------------------------------
// Generic grid-stride fallback (any row count): direct NT b128 stream.
// ---------------------------------------------------------------------------
__global__ __launch_bounds__(BLOCK) void ortho2_direct(const v4f* __restrict__ in,
                                                       v4f* __restrict__ out,
                                                       long rows) {
  size_t i = (size_t)blockIdx.x * BLOCK + threadIdx.x;
  const size_t stride = (size_t)gridDim.x * BLOCK;
  for (; i < (size_t)rows; i += stride) {
    const v4f v = __builtin_nontemporal_load(&in[i]);
    __builtin_nontemporal_store(gram2(v), &out[i]);
  }
}

extern "C" void kernel_launch(void* const* d_in, const int* in_sizes, int n_in,
                              void* d_out, int out_size, void* d_ws,
                              size_t ws_size, hipStream_t stream) {
  (void)n_in;
  (void)out_size;
  (void)d_ws;
  (void)ws_size;

  const v4f* in = (const v4f*)d_in[0];
  v4f* out = (v4f*)d_out;
  const long rows = (long)in_sizes[0] / 4;  // (N,4) float32 rows

  const long T = 1L << 20;  // 4096 blocks * 256 threads
  if (rows >= T && (rows % T) == 0 && (rows / T) <= 1024) {
    ortho2_pipe<<<(int)(T / BLOCK), BLOCK, 0, stream>>>(in, out, (int)T,
                                                        (int)(rows / T));
  } else {
    long blocks = (rows + BLOCK - 1) / BLOCK;
    if (blocks > 16384) blocks = 16384;
    if (blocks < 1) blocks = 1;
    ortho2_direct<<<(int)blocks, BLOCK, 0, stream>>>(in, out, rows);
  }
}